// SKA_65180423685548
// MI455X (gfx1250) — compile-verified
//
#include <hip/hip_runtime.h>
#include <hip/hip_bf16.h>

// SKA / involution-style per-pixel 3x3 grouped conv.
// x: (4,256,96,96) f32, w: (4,32,9,96,96) f32, groups=8, pad=1.
// out[b,c,r,col] = sum_{ky,kx} xpad[b,c,r+ky,col+kx] * w[b,c/8,3ky+kx,r,col]
//
// Memory-bound (170 MFLOP vs ~120 MB -> ~5us at 23.3 TB/s), so the MI455X
// feature that matters is the async global->LDS pipeline (ASYNCcnt), not WMMA.

#define TILE_ROWS 8
#define LDS_ROWS 10          // TILE_ROWS + 2 halo rows
#define LDS_STRIDE 104       // floats per LDS row; interior cols start at idx 4 (16B aligned)
#define PLANE 9216           // 96*96

__device__ __forceinline__ void async_tile_load_b128(unsigned ldsByteOff,
                                                     unsigned gByteOff,
                                                     const float* saddr) {
    // CDNA5 async copy: copies 16 bytes/lane from global saddr + gByteOff to LDS offset ldsByteOff.
    // Tracked with ASYNCcnt (not LOADcnt).
    asm volatile("global_load_async_to_lds_b128 %0, %1, %2"
                 :
                 : "v"(ldsByteOff), "v"(gByteOff), "s"(saddr)
                 : "memory");
}

__global__ __launch_bounds__(256)
void SKA_65180423685548_kernel(const float* __restrict__ x,
                               const float* __restrict__ w,
                               float* __restrict__ out) {
    __shared__ float lds[2][LDS_ROWS * LDS_STRIDE];   // 8320 B

    const int tid = threadIdx.x;
    const int rb  = blockIdx.x;   // 12 row blocks of 8 rows
    const int gc  = blockIdx.y;   // 32 weight-channel groups
    const int b   = blockIdx.z;   // 4 batches
    const int r0  = rb * TILE_ROWS;

    // Pre-zero both LDS buffers with b128 stores: halo columns and
    // out-of-image halo rows stay zero forever (the async-load skip pattern
    // is identical every tile, so skipped slots are never overwritten).
    {
        float4* p = (float4*)&lds[0][0];
        const float4 z = make_float4(0.f, 0.f, 0.f, 0.f);
#pragma unroll
        for (int i = 0; i < 2; ++i)
            p[tid + 256 * i] = z;                     // 512 of 520
        if (tid < 2 * LDS_ROWS * LDS_STRIDE / 4 - 512)
            p[tid + 512] = z;                         // remaining 8
    }
    __syncthreads();

    // ---- async x-tile loader mapping: 240 b128 loads, one per thread ----
    const int  trow   = tid / 24;               // tile row 0..9
    const int  tc4    = tid - trow * 24;        // float4 column 0..23
    const int  grow   = r0 - 1 + trow;          // global row (-1..96)
    const bool doload = (tid < 240) && (grow >= 0) && (grow < 96);
    const unsigned voff = (unsigned)((grow * 96 + tc4 * 4) * 4);
    const unsigned ldsOff0 =
        (unsigned)(uintptr_t)&lds[0][trow * LDS_STRIDE + 4 + tc4 * 4];
    const unsigned ldsOff1 =
        (unsigned)(uintptr_t)&lds[1][trow * LDS_STRIDE + 4 + tc4 * 4];

    const float* xg = x + (size_t)(b * 256 + gc * 8) * PLANE;  // channel g=0 plane

    // Prime the pipeline: channel 0 -> buffer 0.
    if (doload) async_tile_load_b128(ldsOff0, voff, xg);

    // ---- per-thread weights: 9 taps x 4 pixels, reused for all 8 channels ----
    const bool docomp = tid < 192;              // 8 rows x 24 float4
    const int  orow   = tid / 24;               // output row in strip (when docomp)
    const int  oc4    = (tid - (tid / 24) * 24) * 4;  // output col
    float4 wq[9];
    float* outp = nullptr;
    if (docomp) {
        const float* wbase = w + (size_t)(b * 32 + gc) * 9 * PLANE
                               + (r0 + orow) * 96 + oc4;
#pragma unroll
        for (int k = 0; k < 9; ++k)
            wq[k] = *(const float4*)(wbase + k * PLANE);
        outp = out + (size_t)(b * 256 + gc * 8) * PLANE + (r0 + orow) * 96 + oc4;
    }

#pragma unroll
    for (int g = 0; g < 8; ++g) {
        // Issue next channel's tile into the other buffer (its previous readers
        // all passed the trailing barrier of iteration g-1), then wait for the
        // current tile only: async loads complete in order, so asynccnt<=1
        // guarantees tile g has landed while tile g+1 stays in flight.
        if (g + 1 < 8) {
            if (doload)
                async_tile_load_b128((g & 1) ? ldsOff0 : ldsOff1, voff,
                                     xg + (size_t)(g + 1) * PLANE);
            asm volatile("s_wait_asynccnt 1" ::: "memory");
        } else {
            asm volatile("s_wait_asynccnt 0" ::: "memory");
        }
        __syncthreads();

        if (docomp) {
            const float* buf = lds[g & 1];
            float a0 = 0.f, a1 = 0.f, a2 = 0.f, a3 = 0.f;
#pragma unroll
            for (int ky = 0; ky < 3; ++ky) {
                const float* row = buf + (orow + ky) * LDS_STRIDE + oc4 + 3;
                const float x0 = row[0], x1 = row[1], x2 = row[2];
                const float x3 = row[3], x4 = row[4], x5 = row[5];
                const float4 w0 = wq[ky * 3 + 0];
                const float4 w1 = wq[ky * 3 + 1];
                const float4 w2 = wq[ky * 3 + 2];
                a0 = fmaf(x0, w0.x, fmaf(x1, w1.x, fmaf(x2, w2.x, a0)));
                a1 = fmaf(x1, w0.y, fmaf(x2, w1.y, fmaf(x3, w2.y, a1)));
                a2 = fmaf(x2, w0.z, fmaf(x3, w1.z, fmaf(x4, w2.z, a2)));
                a3 = fmaf(x3, w0.w, fmaf(x4, w1.w, fmaf(x5, w2.w, a3)));
            }
            *(float4*)(outp + (size_t)g * PLANE) = make_float4(a0, a1, a2, a3);
        }
        __syncthreads();  // protect buffer (g&1) before it is refilled at g+2
    }
}

extern "C" void kernel_launch(void* const* d_in, const int* in_sizes, int n_in,
                              void* d_out, int out_size, void* d_ws, size_t ws_size,
                              hipStream_t stream) {
    const float* x = (const float*)d_in[0];   // 4*256*96*96
    const float* w = (const float*)d_in[1];   // 4*32*9*96*96
    float* out = (float*)d_out;               // 4*256*96*96

    dim3 grid(96 / TILE_ROWS, 32, 4);         // (12, 32, 4)
    dim3 block(256);
    SKA_65180423685548_kernel<<<grid, block, 0, stream>>>(x, w, out);
}